// LearningModel_10247791968674
// MI455X (gfx1250) — compile-verified
//
#include <hip/hip_runtime.h>
#include <hip/hip_bf16.h>

typedef __attribute__((ext_vector_type(2))) float v2f;
typedef __attribute__((ext_vector_type(8))) float v8f;

#define DIM      128     // embedding dim
#define KDIM     256     // 2*DIM (concat of two parents)
#define MPL      4096    // derived nodes per level
#define NRULES   256
#define TSTRIDE  260     // LDS row stride in floats (bank-conflict-free b64 reads)
#define RED_BLOCKS 512

__device__ __forceinline__ float softplusf(float x) {
  return fmaxf(x, 0.0f) + log1pf(expf(-fabsf(x)));
}

// ---------------------------------------------------------------------------
// store[0:n_init] = thax_table[init_thax] + sine_table[init_sine]
// ---------------------------------------------------------------------------
__global__ __launch_bounds__(256) void init_embed_kernel(
    float* __restrict__ store,
    const float* __restrict__ thax_t, const float* __restrict__ sine_t,
    const int* __restrict__ init_thax, const int* __restrict__ init_sine,
    int n_init) {
  int idx = blockIdx.x * blockDim.x + threadIdx.x;
  int total = n_init * (DIM / 4);
  if (idx >= total) return;
  int node = idx >> 5;
  int seg  = idx & 31;
  float4 a = ((const float4*)(thax_t + (size_t)init_thax[node] * DIM))[seg];
  float4 b = ((const float4*)(sine_t + (size_t)init_sine[node] * DIM))[seg];
  float4 o;
  o.x = a.x + b.x; o.y = a.y + b.y; o.z = a.z + b.z; o.w = a.w + b.w;
  ((float4*)(store + (size_t)node * DIM))[seg] = o;
}

// ---------------------------------------------------------------------------
// Pre-swizzle rule_W into per-lane WMMA B-fragment order:
//   Bp[r][wv(8)][ks(64)][lane(32)][2] = W[r][4*ks + 2*(lane>>4) + {0,1}][wv*16 + (lane&15)]
// -> inner loop B fetch = one coalesced global_load_b64 per K-step.
// ---------------------------------------------------------------------------
__global__ __launch_bounds__(256) void pack_b_kernel(
    const float* __restrict__ rule_W, float* __restrict__ Bp) {
  int u = blockIdx.x * 256 + threadIdx.x;      // [0, NRULES*8*64*32)
  int lane = u & 31;
  int ks   = (u >> 5) & 63;
  int wv   = (u >> 11) & 7;
  int r    = u >> 14;
  int half = lane >> 4, lrow = lane & 15;
  size_t srow = ((size_t)r * KDIM + 4 * ks + 2 * half) * DIM + wv * 16 + lrow;
  float2 v;
  v.x = rule_W[srow];
  v.y = rule_W[srow + DIM];
  ((float2*)Bp)[u] = v;
}

// ---------------------------------------------------------------------------
// Per-level rule binning (block = level). In-rule permutation is value-neutral.
// ---------------------------------------------------------------------------
__global__ __launch_bounds__(256) void bin_all_kernel(
    const int* __restrict__ rules,        // [L, MPL]
    int* __restrict__ rule_off_all,       // [L, NRULES+1]
    int* __restrict__ sorted_all) {       // [L, MPL]
  __shared__ int cnt[NRULES];
  __shared__ int off[NRULES + 1];
  __shared__ int cur[NRULES];
  const int l   = blockIdx.x;
  const int tid = threadIdx.x;
  const int* rl = rules + (size_t)l * MPL;
  int* offg     = rule_off_all + (size_t)l * (NRULES + 1);
  int* sorted   = sorted_all + (size_t)l * MPL;

  cnt[tid] = 0;
  __syncthreads();
  for (int i = tid; i < MPL; i += 256) atomicAdd(&cnt[rl[i]], 1);
  __syncthreads();
  if (tid == 0) {
    int acc = 0;
    for (int r = 0; r < NRULES; ++r) { off[r] = acc; acc += cnt[r]; }
    off[NRULES] = acc;
  }
  __syncthreads();
  cur[tid] = off[tid];
  __syncthreads();
  for (int i = tid; i < MPL; i += 256) {
    int r = rl[i];
    int p = atomicAdd(&cur[r], 1);
    sorted[p] = i;
  }
  offg[tid] = off[tid];
  if (tid == 0) offg[NRULES] = off[NRULES];
}

// ---------------------------------------------------------------------------
// One level. Block = rule r; 8 waves, wave wv owns N-tile [16*wv,16*wv+16).
// K=256 accumulated with V_WMMA_F32_16X16X4_F32 (full fp32 precision).
// C goes back through LDS -> coalesced b128 row stores.
// ---------------------------------------------------------------------------
template <bool PACKED>
__global__ __launch_bounds__(256) void gemm_level_kernel(
    float* __restrict__ store,
    const float* __restrict__ rule_W,     // [NRULES, KDIM, DIM]
    const float* __restrict__ Bp,         // packed fragments (if PACKED)
    const float* __restrict__ rule_b,     // [NRULES, DIM]
    const int* __restrict__ parents_l,    // [MPL, 2]
    const int* __restrict__ sorted_idx,   // [MPL] rule-sorted node ids
    const int* __restrict__ rule_off,     // [NRULES+1]
    int out_base) {                       // n_init + l*MPL
  __shared__ __align__(16) float Atile[16 * TSTRIDE];
  __shared__ int nid[16];

  const int r   = blockIdx.x;
  const int s0  = rule_off[r];
  const int s1  = rule_off[r + 1];
  const int tid = threadIdx.x;
  const int ntiles = (s1 - s0 + 15) >> 4;

  const int wv   = tid >> 5;   // wave 0..7
  const int lane = tid & 31;
  const int half = lane >> 4;  // 0: K+{0,1}, C rows 0..7 ; 1: K+{2,3}, C rows 8..15
  const int lrow = lane & 15;  // A row (M) / B column (N) within tile

  const int row = tid >> 4;    // staging / writeback row 0..15
  const int seg = tid & 15;

  const float bias = rule_b[r * DIM + wv * 16 + lrow];

  for (int t = 0; t < ntiles; ++t) {
    // ---- stage A tile: 16 rows x 256 floats (two gathered parent rows) ----
    {
      int srcNode = -1;
      const int slot = s0 + t * 16 + row;
      if (slot < s1) srcNode = sorted_idx[slot];
      if (seg == 0) nid[row] = srcNode;
      float4 v0 = {0,0,0,0}, v1 = v0, v2 = v0, v3 = v0;
      if (srcNode >= 0) {
        const int par = parents_l[2 * srcNode + (seg >> 3)];
        const float4* src = (const float4*)(store + (size_t)par * DIM + (seg & 7) * 16);
        v0 = src[0]; v1 = src[1]; v2 = src[2]; v3 = src[3];
      }
      float4* dst = (float4*)(Atile + row * TSTRIDE + seg * 16);
      dst[0] = v0; dst[1] = v1; dst[2] = v2; dst[3] = v3;
    }
    __syncthreads();

    // ---- 16x16 fp32 WMMA, 64 K-steps of 4 ----
    v8f c = {0.f,0.f,0.f,0.f,0.f,0.f,0.f,0.f};
    const float* Arow = Atile + lrow * TSTRIDE + 2 * half;   // 8B aligned
    const float* Bq  = PACKED
        ? Bp + ((size_t)(r * 8 + wv) * 64) * 64 + lane * 2
        : rule_W + ((size_t)r * KDIM + 2 * half) * DIM + (wv * 16 + lrow);
    #pragma unroll 8
    for (int ks = 0; ks < 64; ++ks) {
      v2f a = *(const v2f*)(Arow + 4 * ks);
      v2f b;
      if (PACKED) {
        b = *(const v2f*)(Bq + ks * 64);          // 256B/wave coalesced b64
      } else {
        b.x = Bq[ks * 4 * DIM];                   // B[4ks+2h  ][16wv+lrow]
        b.y = Bq[ks * 4 * DIM + DIM];             // B[4ks+2h+1][16wv+lrow]
      }
      c = __builtin_amdgcn_wmma_f32_16x16x4_f32(
              false, a, false, b, (short)0, c, false, false);
    }

    // ---- bias + relu, transpose C through LDS ----
    __syncthreads();                              // all A reads done
    #pragma unroll
    for (int v = 0; v < 8; ++v) {
      float val = c[v] + bias;
      val = val > 0.f ? val : 0.f;
      Atile[(v + 8 * half) * TSTRIDE + wv * 16 + lrow] = val;
    }
    __syncthreads();

    // ---- coalesced row writeback: 2 x global_store_b128 per thread ----
    {
      const int node = nid[row];
      if (node >= 0) {
        const float4* srcp = (const float4*)(Atile + row * TSTRIDE + seg * 8);
        float4* dst = (float4*)(store + (size_t)(out_base + node) * DIM + seg * 8);
        dst[0] = srcp[0];
        dst[1] = srcp[1];
      }
    }
    __syncthreads();
  }
}

// ---------------------------------------------------------------------------
// Stage 1 reduction (deterministic; no float atomics).
// ---------------------------------------------------------------------------
__global__ __launch_bounds__(256) void reduce1_kernel(
    const float* __restrict__ store,
    const float* __restrict__ eval_w, const float* __restrict__ eval_b,
    const float* __restrict__ pos_vals, const float* __restrict__ neg_vals,
    int n_tot, float* __restrict__ partials) {  // [gridDim.x, 6]
  __shared__ float acc[8][6];
  const int wv = threadIdx.x >> 5, lane = threadIdx.x & 31;
  const float4 w4 = ((const float4*)eval_w)[lane];
  const float eb = eval_b[0];
  float sA = 0.f, sB = 0.f, sP = 0.f, sN = 0.f, sPOK = 0.f, sNOK = 0.f;
  const int gw = blockIdx.x * 8 + wv;
  const int stride = gridDim.x * 8;
  for (int rowi = gw; rowi < n_tot; rowi += stride) {
    float4 x = ((const float4*)(store + (size_t)rowi * DIM))[lane];
    float d = x.x * w4.x + x.y * w4.y + x.z * w4.z + x.w * w4.w;
    #pragma unroll
    for (int m = 16; m >= 1; m >>= 1) d += __shfl_xor(d, m, 32);
    if (lane == 0) {
      float logit = d + eb;
      float pos = pos_vals[rowi], neg = neg_vals[rowi];
      sA += pos * softplusf(-logit);
      sB += neg * softplusf(logit);
      sP += pos; sN += neg;
      if (logit >= 0.f) sPOK += pos; else sNOK += neg;
    }
  }
  if (lane == 0) {
    acc[wv][0] = sA; acc[wv][1] = sB; acc[wv][2] = sP;
    acc[wv][3] = sN; acc[wv][4] = sPOK; acc[wv][5] = sNOK;
  }
  __syncthreads();
  if (threadIdx.x < 6) {
    float s = 0.f;
    for (int w = 0; w < 8; ++w) s += acc[w][threadIdx.x];
    partials[blockIdx.x * 6 + threadIdx.x] = s;
  }
}

__global__ void reduce2_kernel(const float* __restrict__ partials, int nblocks,
                               float* __restrict__ out) {
  __shared__ float tot[6];
  if (threadIdx.x < 6) {
    float s = 0.f;
    for (int b = 0; b < nblocks; ++b) s += partials[b * 6 + threadIdx.x];
    tot[threadIdx.x] = s;
  }
  __syncthreads();
  if (threadIdx.x == 0) {
    float pw = tot[3] / tot[2];            // sum(neg) / sum(pos)
    out[0] = pw * tot[0] + tot[1];         // loss
    out[1] = tot[4];                       // posOK
    out[2] = tot[5];                       // negOK
  }
}

// ---------------------------------------------------------------------------
extern "C" void kernel_launch(void* const* d_in, const int* in_sizes, int n_in,
                              void* d_out, int out_size, void* d_ws, size_t ws_size,
                              hipStream_t stream) {
  const float* thax_table = (const float*)d_in[0];
  const float* sine_table = (const float*)d_in[1];
  const float* rule_W     = (const float*)d_in[2];
  const float* rule_b     = (const float*)d_in[3];
  const float* eval_w     = (const float*)d_in[4];
  const float* eval_b     = (const float*)d_in[5];
  const float* pos_vals   = (const float*)d_in[6];
  const float* neg_vals   = (const float*)d_in[7];
  const int*   init_thax  = (const int*)d_in[8];
  const int*   init_sine  = (const int*)d_in[9];
  const int*   parents    = (const int*)d_in[10];   // [L, MPL, 2]
  const int*   rules      = (const int*)d_in[11];   // [L, MPL]

  const int n_init   = in_sizes[8];
  const int n_tot    = in_sizes[6];
  const int n_levels = in_sizes[11] / MPL;

  // workspace layout (Bp last so the base layout is packing-independent)
  char* ws = (char*)d_ws;
  size_t off = 0;
  float* store = (float*)(ws + off);
  off += (size_t)n_tot * DIM * sizeof(float);
  off = (off + 255) & ~(size_t)255;
  int* rule_off_all = (int*)(ws + off);
  off += (size_t)n_levels * (NRULES + 1) * sizeof(int);
  off = (off + 255) & ~(size_t)255;
  int* sorted_all = (int*)(ws + off);
  off += (size_t)n_levels * MPL * sizeof(int);
  off = (off + 255) & ~(size_t)255;
  float* partials = (float*)(ws + off);
  off += (size_t)RED_BLOCKS * 6 * sizeof(float);
  off = (off + 255) & ~(size_t)255;
  float* Bp = (float*)(ws + off);
  const size_t pack_bytes = (size_t)NRULES * KDIM * DIM * sizeof(float);
  const bool packed = (off + pack_bytes) <= ws_size;
  (void)n_in; (void)out_size;

  // 1) axiom embeddings
  {
    int total = n_init * (DIM / 4);
    init_embed_kernel<<<(total + 255) / 256, 256, 0, stream>>>(
        store, thax_table, sine_table, init_thax, init_sine, n_init);
  }

  // 2) one-time B fragment swizzle (reused by all 64 levels out of L2)
  if (packed) {
    pack_b_kernel<<<(NRULES * 8 * 64 * 32) / 256, 256, 0, stream>>>(rule_W, Bp);
  }

  // 3) bin every level's nodes by rule
  bin_all_kernel<<<n_levels, 256, 0, stream>>>(rules, rule_off_all, sorted_all);

  // 4) derived levels, one kernel per level (stream order = level barrier)
  for (int l = 0; l < n_levels; ++l) {
    if (packed) {
      gemm_level_kernel<true><<<NRULES, 256, 0, stream>>>(
          store, rule_W, Bp, rule_b,
          parents + (size_t)l * MPL * 2,
          sorted_all + (size_t)l * MPL,
          rule_off_all + (size_t)l * (NRULES + 1),
          n_init + l * MPL);
    } else {
      gemm_level_kernel<false><<<NRULES, 256, 0, stream>>>(
          store, rule_W, Bp, rule_b,
          parents + (size_t)l * MPL * 2,
          sorted_all + (size_t)l * MPL,
          rule_off_all + (size_t)l * (NRULES + 1),
          n_init + l * MPL);
    }
  }

  // 5) deterministic two-stage reduction
  reduce1_kernel<<<RED_BLOCKS, 256, 0, stream>>>(
      store, eval_w, eval_b, pos_vals, neg_vals, n_tot, partials);
  reduce2_kernel<<<1, 64, 0, stream>>>(partials, RED_BLOCKS, (float*)d_out);
}